// Agent_62594853372414
// MI455X (gfx1250) — compile-verified
//
#include <hip/hip_runtime.h>
#include <hip/hip_bf16.h>

// CDNA5 / gfx1250, wave32. All GEMMs via V_WMMA_F32_16X16X4_F32 with the
// gathered A-tile staged in LDS (one block = one M-tile, 8 waves = 8 N-tiles).

typedef __attribute__((ext_vector_type(2))) float v2f;
typedef __attribute__((ext_vector_type(8))) float v8f;

#define KMAX 256
#define ROWPAD 4            // LDS row stride = K+4 floats: 16B-aligned rows, lane m -> bank 4m

// ---------------------------------------------------------------------------
// Generic gathered-row fp32 WMMA GEMM:
//   out[r][n] = act( A[src(r)][:K] . W[n][wOff..wOff+K] + bias[n]
//                    + add1[add1Idx(r)][n] + add2[r][n] )
// grid.x = Mtiles; block = 256 (8 waves). Wave w computes N-tile w (Nout=128).
// The 16 gathered A rows are staged once per block into LDS.
// ---------------------------------------------------------------------------
__launch_bounds__(256)
__global__ void wmma_gemm_f32(const float* __restrict__ A, int lda,
                              const int* __restrict__ rowIdx, int srcClamp,
                              int M,
                              const float* __restrict__ W, int ldw, int wOff, int K,
                              const float* __restrict__ bias,
                              const float* __restrict__ add1,
                              const int* __restrict__ add1Idx, int ldAdd1,
                              const float* __restrict__ add2, int ldAdd2,
                              float* __restrict__ out, int ldo, int applyLeaky)
{
    __shared__ float tileA[16 * (KMAX + ROWPAD)];
    const int mt  = blockIdx.x;
    const int tid = threadIdx.x;
    const int rowStride = K + ROWPAD;

    // Cooperative stage of 16 gathered A rows (K floats each) into LDS, float4 chunks.
    const int vecs = (16 * K) >> 2;
    for (int i = tid; i < vecs; i += 256) {
        int e   = i << 2;                 // flat float index in 16xK tile
        int row = e / K;
        int k   = e - row * K;
        int r   = mt * 16 + row;
        int rc  = (r < M) ? r : (M - 1);
        int src = rowIdx ? rowIdx[rc] : ((rc < srcClamp) ? rc : (srcClamp - 1));
        float4 v = *(const float4*)(A + (size_t)src * lda + k);
        *(float4*)(&tileA[row * rowStride + k]) = v;   // row start 16B aligned (K+4 stride)
    }
    __syncthreads();

    const int wave  = tid >> 5;           // = N-tile index (Ntiles == 8)
    const int lane  = tid & 31;
    const int nt    = wave;
    const int laneM = lane & 15;
    const int kh    = (lane >> 4) << 1;   // 0 or 2 (K-half select)

    // B = W^T : lane n = laneM reads 2 consecutive K values from weight row n
    const float* wrow = W + (size_t)(nt * 16 + laneM) * ldw + wOff + kh;
    const float* asrc = &tileA[laneM * rowStride + kh];
    __builtin_prefetch(wrow, 0, 3);       // global_prefetch_b8

    v8f c = {};
    for (int k = 0; k < K; k += 4) {
        float2 av = *(const float2*)(asrc + k);   // ds_load_b64, conflict-free
        float2 bv = *(const float2*)(wrow + k);   // global_load_b64
        v2f a; a[0] = av.x; a[1] = av.y;
        v2f b; b[0] = bv.x; b[1] = bv.y;
        c = __builtin_amdgcn_wmma_f32_16x16x4_f32(false, a, false, b,
                                                  (short)0, c, false, false);
    }

    // C/D layout: VGPR v -> row (v + 8*(lane>>4)), col (lane & 15)
    const int col   = nt * 16 + (lane & 15);
    const int rbase = mt * 16 + ((lane >> 4) << 3);
    const float bval = bias ? bias[col] : 0.0f;
    #pragma unroll
    for (int v = 0; v < 8; ++v) {
        int row = rbase + v;
        if (row < M) {
            float x = c[v] + bval;
            if (add1) {
                int ai = add1Idx ? add1Idx[row] : row;
                x += add1[(size_t)ai * ldAdd1 + col];
            }
            if (add2) x += add2[(size_t)row * ldAdd2 + col];
            if (applyLeaky) x = (x >= 0.0f) ? x : 0.01f * x;
            out[(size_t)row * ldo + col] = x;
        }
    }
}

// ---------------------------------------------------------------------------
// Neighbor aggregation: agg[b,t,:] = (cnt0*node0p[b] + sum_valid relP[rel] + nn*bp)/max(nn,1)
// grid = B*TOPK, block = 128 (one column per thread). relP (256KB) lives in L2.
// ---------------------------------------------------------------------------
__global__ void agg_kernel(const int* __restrict__ nbn, const int* __restrict__ nbr,
                           const int* __restrict__ nnum,
                           const float* __restrict__ relP,
                           const float* __restrict__ node0p,
                           const float* __restrict__ bp,
                           float* __restrict__ agg)
{
    const int pair = blockIdx.x;
    const int b    = pair >> 5;
    const int col  = threadIdx.x;
    const int nn   = nnum[pair];
    const int* nb = nbn + (size_t)pair * 64;
    const int* nr = nbr + (size_t)pair * 64;
    float acc = 0.0f; int cnt0 = 0;
    for (int m = 0; m < nn; ++m) {
        acc  += relP[(size_t)nr[m] * 128 + col];
        cnt0 += (nb[m] == 0);
    }
    const float denom = (nn > 0) ? (float)nn : 1.0f;
    agg[(size_t)pair * 128 + col] =
        (acc + (float)cnt0 * node0p[(size_t)b * 128 + col] + (float)nn * bp[col]) / denom;
}

// ---------------------------------------------------------------------------
// Sparse node_emb lookup (post-scatter, last-write-wins):
//   out[r][0:128] = upd[b, last t: node_pos==n] | node0[b] (n==0) | 0
//   optional: out[r][128:256] = extra[b]  (query concat for `cur`)
// ---------------------------------------------------------------------------
__global__ void node_lookup_kernel(const int* __restrict__ idx, int perB,
                                   const int* __restrict__ node_pos,
                                   const float* __restrict__ node0,
                                   const float* __restrict__ upd,
                                   const float* __restrict__ extra,
                                   float* __restrict__ out, int ldo)
{
    const int r = blockIdx.x;
    const int b = r / perB;
    const int col = threadIdx.x;
    const int n = idx[r];
    int found = -1;
    #pragma unroll
    for (int t = 0; t < 32; ++t)
        if (node_pos[b * 32 + t] == n) found = t;   // last write wins
    float v;
    if (found >= 0)      v = upd[((size_t)(b * 32 + found)) * 128 + col];
    else if (n == 0)     v = node0[(size_t)b * 128 + col];
    else                 v = 0.0f;
    out[(size_t)r * ldo + col] = v;
    if (extra) out[(size_t)r * ldo + 128 + col] = extra[(size_t)b * 128 + col];
}

// thresholds[b] = cur[b] . Wg + bg   -> d_out[b*65 + 64]
__global__ void thresh_kernel(const float* __restrict__ cur, const float* __restrict__ Wg,
                              const float* __restrict__ bg, float* __restrict__ out)
{
    __shared__ float sm[256];
    const int b = blockIdx.x, t = threadIdx.x;
    sm[t] = cur[(size_t)b * 256 + t] * Wg[t];
    __syncthreads();
    for (int s = 128; s > 0; s >>= 1) {
        if (t < s) sm[t] += sm[t + s];
        __syncthreads();
    }
    if (t == 0) out[b * 65 + 64] = sm[0] + bg[0];
}

// scores[b][m] = mask ? (state[b].cand[b,m])/sqrt(128) : -1e5  -> d_out[b*65 + m]
__global__ void scores_kernel(const float* __restrict__ state, const float* __restrict__ cand,
                              const int* __restrict__ mask, float* __restrict__ out)
{
    const int b = blockIdx.x, m = threadIdx.x;
    const float* s = state + (size_t)b * 128;
    const float* c = cand + ((size_t)b * 64 + m) * 128;
    float d = 0.0f;
    for (int h = 0; h < 128; ++h) d += s[h] * c[h];
    d *= 0.08838834764831845f;   // 1/sqrt(128)
    out[b * 65 + m] = mask[b * 64 + m] ? d : -100000.0f;
}

// node_scores[b][n] = base[b] (+ node-row . Wr[:128] at sparse positions)
__global__ void node_scores_kernel(const float* __restrict__ query,
                                   const int* __restrict__ node_pos,
                                   const float* __restrict__ node0,
                                   const float* __restrict__ upd,
                                   const float* __restrict__ Wr,
                                   const float* __restrict__ br,
                                   float* __restrict__ out_ns)
{
    const int b = blockIdx.x;
    const int tid = threadIdx.x, lane = tid & 31, wave = tid >> 5;
    __shared__ float s_base;
    if (wave == 0) {
        float p = 0.0f;
        #pragma unroll
        for (int j = 0; j < 4; ++j) { int k = lane * 4 + j; p += query[(size_t)b * 128 + k] * Wr[128 + k]; }
        #pragma unroll
        for (int off = 16; off > 0; off >>= 1) p += __shfl_xor(p, off, 32);
        if (lane == 0) s_base = p + br[0];
    }
    __syncthreads();
    const float base = s_base;
    for (int n = tid; n < 1024; n += 256) out_ns[(size_t)b * 1024 + n] = base;
    __syncthreads();
    // scattered corrections: wave w handles t in [4w, 4w+4)
    for (int i = 0; i < 4; ++i) {
        const int t = wave * 4 + i;
        const int n = node_pos[b * 32 + t];
        bool last = true;
        for (int tt = t + 1; tt < 32; ++tt)
            if (node_pos[b * 32 + tt] == n) last = false;
        float p = 0.0f;
        #pragma unroll
        for (int j = 0; j < 4; ++j) { int k = lane * 4 + j; p += upd[((size_t)(b * 32 + t)) * 128 + k] * Wr[k]; }
        #pragma unroll
        for (int off = 16; off > 0; off >>= 1) p += __shfl_xor(p, off, 32);
        if (lane == 0 && last && n < 1024) out_ns[(size_t)b * 1024 + n] = base + p;
    }
    // node 0 (only if not overwritten by scatter)
    if (wave == 0) {
        bool ow = false;
        for (int t = 0; t < 32; ++t) if (node_pos[b * 32 + t] == 0) ow = true;
        if (!ow) {
            float p = 0.0f;
            #pragma unroll
            for (int j = 0; j < 4; ++j) { int k = lane * 4 + j; p += node0[(size_t)b * 128 + k] * Wr[k]; }
            #pragma unroll
            for (int off = 16; off > 0; off >>= 1) p += __shfl_xor(p, off, 32);
            if (lane == 0) out_ns[(size_t)b * 1024] = base + p;
        }
    }
}

// ---------------------------------------------------------------------------
extern "C" void kernel_launch(void* const* d_in, const int* in_sizes, int n_in,
                              void* d_out, int out_size, void* d_ws, size_t ws_size,
                              hipStream_t stream) {
    (void)in_sizes; (void)n_in; (void)out_size; (void)ws_size;
    const int*   start_entities      = (const int*)d_in[0];
    const float* query               = (const float*)d_in[1];
    const int*   node_pos            = (const int*)d_in[2];
    const int*   aims                = (const int*)d_in[3];
    const int*   neighbor_nodes     = (const int*)d_in[4];
    const int*   neighbor_relations = (const int*)d_in[5];
    const int*   neighbors_num      = (const int*)d_in[6];
    const int*   currents            = (const int*)d_in[7];
    const int*   candidate_nodes     = (const int*)d_in[8];
    const int*   candidate_entities  = (const int*)d_in[9];
    const int*   candidate_relations = (const int*)d_in[10];
    const int*   candidate_masks     = (const int*)d_in[11];
    const float* entity_emb   = (const float*)d_in[12];
    const float* relation_emb = (const float*)d_in[13];
    const float* Wh  = (const float*)d_in[14];
    const float* bh  = (const float*)d_in[15];
    const float* Wp  = (const float*)d_in[16];
    const float* bp  = (const float*)d_in[17];
    const float* Wnh = (const float*)d_in[18];
    const float* bnh = (const float*)d_in[19];
    const float* Wc  = (const float*)d_in[20];
    const float* bc  = (const float*)d_in[21];
    const float* Wg  = (const float*)d_in[22];
    const float* bg  = (const float*)d_in[23];
    const float* Wr  = (const float*)d_in[24];
    const float* br  = (const float*)d_in[25];

    float* out = (float*)d_out;
    float* ws  = (float*)d_ws;

    // workspace layout (floats)
    float* f_node0   = ws;             // 256*128
    float* f_node0p  = ws + 32768;     // 256*128
    float* f_relP    = ws + 65536;     // 512*128 (500 used)
    float* f_relWc   = ws + 131072;    // 512*128
    float* f_agg     = ws + 196608;    // 8192*128
    float* f_upd     = ws + 1245184;   // 8192*128
    float* f_cur     = ws + 2293760;   // 256*256
    float* f_state   = ws + 2359296;   // 256*128
    float* f_candN   = ws + 2392064;   // 16384*128
    float* f_candAcc = ws + 4489216;   // 16384*128
    float* f_cand    = ws + 6586368;   // 16384*128  (total 33.1 MiB)

    auto gemm = [&](const float* A, int lda, const int* rowIdx, int srcClamp, int M,
                    const float* W, int ldw, int wOff, int K,
                    const float* bias,
                    const float* add1, const int* add1Idx, int ldAdd1,
                    const float* add2, int ldAdd2,
                    float* O, int ldo, int leaky) {
        const int Mtiles = (M + 15) / 16;   // one block per M-tile, 8 waves = 8 N-tiles
        wmma_gemm_f32<<<Mtiles, 256, 0, stream>>>(
            A, lda, rowIdx, srcClamp, M, W, ldw, wOff, K,
            bias, add1, add1Idx, ldAdd1, add2, ldAdd2, O, ldo, leaky);
    };

    // 1) rel_projP = relation_emb @ Wp[:,128:256]^T
    gemm(relation_emb, 128, nullptr, 500, 500, Wp, 256, 128, 128,
         nullptr, nullptr, nullptr, 0, nullptr, 0, f_relP, 128, 0);
    // 2) relWc = relation_emb @ Wc[:,256:384]^T
    gemm(relation_emb, 128, nullptr, 500, 500, Wc, 384, 256, 128,
         nullptr, nullptr, nullptr, 0, nullptr, 0, f_relWc, 128, 0);
    // 3) node0 = leaky(entity_emb[start] @ Wh^T + bh)
    gemm(entity_emb, 128, start_entities, 0, 256, Wh, 128, 0, 128,
         bh, nullptr, nullptr, 0, nullptr, 0, f_node0, 128, 1);
    // 4) node0_proj = node0 @ Wp[:,0:128]^T
    gemm(f_node0, 128, nullptr, 256, 256, Wp, 256, 0, 128,
         nullptr, nullptr, nullptr, 0, nullptr, 0, f_node0p, 128, 0);
    // 5) neighbor aggregation
    agg_kernel<<<256 * 32, 128, 0, stream>>>(neighbor_nodes, neighbor_relations,
                                             neighbors_num, f_relP, f_node0p, bp, f_agg);
    // 6) upd = leaky(entity_emb[aims] @ Wh^T + bh + agg)
    gemm(entity_emb, 128, aims, 0, 8192, Wh, 128, 0, 128,
         bh, nullptr, nullptr, 0, f_agg, 128, f_upd, 128, 1);
    // 7) cur = [node_emb[b,currents[b]], query[b]]
    node_lookup_kernel<<<256, 128, 0, stream>>>(currents, 1, node_pos, f_node0, f_upd,
                                                query, f_cur, 256);
    // 8) state = leaky(cur @ Wnh^T + bnh)
    gemm(f_cur, 256, nullptr, 256, 256, Wnh, 256, 0, 256,
         bnh, nullptr, nullptr, 0, nullptr, 0, f_state, 128, 1);
    // 9) thresholds -> d_out[:,64]
    thresh_kernel<<<256, 256, 0, stream>>>(f_cur, Wg, bg, out);
    // 10) candidate node rows (post-scatter lookup)
    node_lookup_kernel<<<256 * 64, 128, 0, stream>>>(candidate_nodes, 64, node_pos,
                                                     f_node0, f_upd, nullptr, f_candN, 128);
    // 11) candAcc = entity_emb[cand_ent] @ Wc[:,128:256]^T
    gemm(entity_emb, 128, candidate_entities, 0, 16384, Wc, 384, 128, 128,
         nullptr, nullptr, nullptr, 0, nullptr, 0, f_candAcc, 128, 0);
    // 12) cand = leaky(candN @ Wc[:,0:128]^T + bc + relWc[cand_rel] + candAcc)
    gemm(f_candN, 128, nullptr, 16384, 16384, Wc, 384, 0, 128,
         bc, f_relWc, candidate_relations, 128, f_candAcc, 128, f_cand, 128, 1);
    // 13) scores -> d_out[:, :64]
    scores_kernel<<<256, 64, 0, stream>>>(f_state, f_cand, candidate_masks, out);
    // 14) node_scores -> d_out + 256*65
    node_scores_kernel<<<256, 256, 0, stream>>>(query, node_pos, f_node0, f_upd,
                                                Wr, br, out + 256 * 65);
}